// GCN_N_33062658245474
// MI455X (gfx1250) — compile-verified
//
#include <hip/hip_runtime.h>
#include <hip/hip_bf16.h>

#define N_NODES 100000
#define N_EDGES 1600000
#define IN_DIM 128
#define H1D 64
#define H2D 32
#define OUT_DIM 16

typedef float v2f __attribute__((ext_vector_type(2)));
typedef float v8f __attribute__((ext_vector_type(8)));

// ---------------------------------------------------------------------------
// init: h1 <- b1 broadcast, h2 <- b2 broadcast, pooled <- -FLT_MAX
// ---------------------------------------------------------------------------
__global__ void gcn_init_kernel(float* __restrict__ h1, const float* __restrict__ b1,
                                float* __restrict__ h2, const float* __restrict__ b2,
                                float* __restrict__ pooled) {
    int i = blockIdx.x * blockDim.x + threadIdx.x;
    int stride = gridDim.x * blockDim.x;
    const int total1 = N_NODES * H1D;
    const int total2 = N_NODES * H2D;
    for (int idx = i; idx < total1; idx += stride) h1[idx] = b1[idx & (H1D - 1)];
    for (int idx = i; idx < total2; idx += stride) h2[idx] = b2[idx & (H2D - 1)];
    if (i < H2D) pooled[i] = -3.402823466e38f;
}

// ---------------------------------------------------------------------------
// GEMM via V_WMMA_F32_16X16X4_F32.
// One wave computes a 16-node x NOUT tile of  Y = X @ W^T  (W is [NOUT,K]).
// A-layout (16x4 f32): lane L holds row M=L%16, elems K={0,1}+2*(L>=16).
// B-layout (4x16 f32): lane L holds col N=L%16, rows K={0,1}+2*(L>=16).
// C/D-layout: VGPR r, lane L -> (M = r + 8*(L>=16), N = L%16).
// ---------------------------------------------------------------------------
template <int K, int NOUT>
__global__ void gcn_wmma_gemm(const float* __restrict__ X, const float* __restrict__ W,
                              float* __restrict__ Y, int nTiles) {
    const int wave = blockIdx.x * (blockDim.x >> 5) + (threadIdx.x >> 5);
    if (wave >= nTiles) return;  // wave-uniform: EXEC stays all-ones for WMMA

    const int lane  = threadIdx.x & 31;
    const int row   = lane & 15;          // M for A-frag, N for B-frag
    const int koff  = (lane >> 4) << 1;   // 0 or 2
    constexpr int NT = NOUT / 16;         // N-tiles

    const float* xrow = X + (size_t)(wave * 16 + row) * K + koff;

    v8f acc[NT];
#pragma unroll
    for (int t = 0; t < NT; ++t) acc[t] = (v8f){0.f, 0.f, 0.f, 0.f, 0.f, 0.f, 0.f, 0.f};

#pragma unroll 4
    for (int k0 = 0; k0 < K; k0 += 4) {
        v2f a = *(const v2f*)(xrow + k0);  // A[M][k0+koff .. +1]
#pragma unroll
        for (int t = 0; t < NT; ++t) {
            // B[k][n] = W[n][k],  n = t*16 + row
            v2f b = *(const v2f*)(W + (size_t)(t * 16 + row) * K + k0 + koff);
            acc[t] = __builtin_amdgcn_wmma_f32_16x16x4_f32(
                false, a, false, b, (short)0, acc[t], false, false);
        }
    }

    const int mhalf = (lane >> 4) << 3;  // 0 or 8
#pragma unroll
    for (int t = 0; t < NT; ++t) {
#pragma unroll
        for (int r = 0; r < 8; ++r) {
            Y[(size_t)(wave * 16 + mhalf + r) * NOUT + t * 16 + (lane & 15)] = acc[t][r];
        }
    }
}

// ---------------------------------------------------------------------------
// edge-parallel aggregation: h[src[e]][:] += y[dst[e]][:]
// one thread per (edge, feature); native f32 atomics (global_atomic_add_f32)
// ---------------------------------------------------------------------------
template <int DIM>
__global__ void gcn_agg_kernel(const float* __restrict__ y, const int* __restrict__ src,
                               const int* __restrict__ dst, float* __restrict__ h, int total) {
    int tid = blockIdx.x * blockDim.x + threadIdx.x;
    if (tid >= total) return;
    int e = tid / DIM;
    int j = tid - e * DIM;
    int s = src[e];
    int d = dst[e];
    unsafeAtomicAdd(&h[(size_t)s * DIM + j], y[(size_t)d * DIM + j]);
}

// ---------------------------------------------------------------------------
// global max pool over nodes: pooled[j] = max_i h2[i][j]
// ---------------------------------------------------------------------------
__device__ __forceinline__ void atomicMaxF32(float* addr, float v) {
    // IEEE ordering trick: works across mixed signs
    if (v >= 0.0f)
        atomicMax((int*)addr, __float_as_int(v));
    else
        atomicMin((unsigned int*)addr, __float_as_uint(v));
}

__global__ void gcn_pool_kernel(const float* __restrict__ h2, float* __restrict__ pooled) {
    __shared__ float lds[256];
    const int tid = threadIdx.x;
    const int total = N_NODES * H2D;
    const int stride = gridDim.x * 256;  // multiple of 32 -> column invariant
    float m = -3.402823466e38f;
    for (int idx = blockIdx.x * 256 + tid; idx < total; idx += stride)
        m = fmaxf(m, h2[idx]);
    lds[tid] = m;
    __syncthreads();
    for (int off = 128; off >= 32; off >>= 1) {
        if (tid < off) lds[tid] = fmaxf(lds[tid], lds[tid + off]);
        __syncthreads();
    }
    if (tid < 32) atomicMaxF32(&pooled[tid], lds[tid]);
}

// ---------------------------------------------------------------------------
// final: out[16] = pooled @ Wf^T + bf
// ---------------------------------------------------------------------------
__global__ void gcn_final_kernel(const float* __restrict__ pooled, const float* __restrict__ Wf,
                                 const float* __restrict__ bf, float* __restrict__ out) {
    int t = threadIdx.x;
    if (t < OUT_DIM) {
        float s = bf[t];
#pragma unroll
        for (int j = 0; j < H2D; ++j) s += pooled[j] * Wf[t * H2D + j];
        out[t] = s;
    }
}

// ---------------------------------------------------------------------------
extern "C" void kernel_launch(void* const* d_in, const int* in_sizes, int n_in,
                              void* d_out, int out_size, void* d_ws, size_t ws_size,
                              hipStream_t stream) {
    const float* x   = (const float*)d_in[0];
    const int* eidx  = (const int*)d_in[1];  // [2, N_EDGES]; src row then dst row
    const float* W1  = (const float*)d_in[2];
    const float* b1  = (const float*)d_in[3];
    const float* W2  = (const float*)d_in[4];
    const float* b2  = (const float*)d_in[5];
    const float* Wf  = (const float*)d_in[6];
    const float* bf  = (const float*)d_in[7];
    float* out = (float*)d_out;

    const int* src = eidx;
    const int* dst = eidx + N_EDGES;

    // workspace layout (floats)
    float* ws = (float*)d_ws;
    float* h1     = ws;                          // N*64
    float* y1     = h1 + (size_t)N_NODES * H1D;  // N*64
    float* h2     = y1 + (size_t)N_NODES * H1D;  // N*32
    float* y2     = h2 + (size_t)N_NODES * H2D;  // N*32
    float* pooled = y2 + (size_t)N_NODES * H2D;  // 32

    // 1) init h1 <- b1, h2 <- b2, pooled <- -inf
    gcn_init_kernel<<<2048, 256, 0, stream>>>(h1, b1, h2, b2, pooled);

    // 2) y1 = x @ W1^T   (WMMA f32 16x16x4, K=128, NOUT=64)
    const int nTiles = N_NODES / 16;  // 6250 exactly
    const int wavesPerBlock = 8;      // 256 threads
    int gemmBlocks = (nTiles + wavesPerBlock - 1) / wavesPerBlock;
    gcn_wmma_gemm<IN_DIM, H1D><<<gemmBlocks, 256, 0, stream>>>(x, W1, y1, nTiles);

    // 3) h1 += A @ y1   (edge-parallel scatter-add, 64-dim)
    {
        int total = N_EDGES * H1D;
        gcn_agg_kernel<H1D><<<(total + 255) / 256, 256, 0, stream>>>(y1, src, dst, h1, total);
    }

    // 4) y2 = h1 @ W2^T  (WMMA f32 16x16x4, K=64, NOUT=32)
    gcn_wmma_gemm<H1D, H2D><<<gemmBlocks, 256, 0, stream>>>(h1, W2, y2, nTiles);

    // 5) h2 += A @ y2   (edge-parallel scatter-add, 32-dim)
    {
        int total = N_EDGES * H2D;
        gcn_agg_kernel<H2D><<<(total + 255) / 256, 256, 0, stream>>>(y2, src, dst, h2, total);
    }

    // 6) pooled = columnwise max over nodes
    gcn_pool_kernel<<<512, 256, 0, stream>>>(h2, pooled);

    // 7) out = pooled @ Wf^T + bf
    gcn_final_kernel<<<1, 32, 0, stream>>>(pooled, Wf, bf, out);
}